// CBOW_5205500363137
// MI455X (gfx1250) — compile-verified
//
#include <hip/hip_runtime.h>
#include <cstdint>
#include <cstddef>

// ---------------------------------------------------------------------------
// CBOW negative-sampling loss, MI455X (gfx1250, wave32).
//
// Memory-bound by >50x: HBM floor ~21us (384MB NT mask streams + ~100MB table
// first-touch; the 2x51MB embedding tables stay resident in the 192MB L2).
// Bandwidth-oriented design:
//   - NT loads for the once-read dropout masks (don't evict L2-resident tables)
//   - RT loads for the embedding-table gathers (L2 hits after first touch)
//   - async global->LDS staging (ASYNCcnt) for the 10 gathered negative rows,
//     with pipelined s_wait_asynccnt N waits (async loads complete in order)
//   - all 32-bit scaled offsets -> SADDR + voffset scale_offset addressing,
//     no 64-bit per-lane address arithmetic
// ---------------------------------------------------------------------------

typedef float v4f __attribute__((ext_vector_type(4)));
typedef int   v4i __attribute__((vector_size(16)));   // matches builtin param type

#define CBOW_B 65536u
#define CBOW_K 10
#define CBOW_E 128u
#define WPB    8                  // waves per block (wave32 -> 256 threads)
#define TPB    (WPB * 32)
#define NBLK   (int)(CBOW_B / WPB)  // 8192 partials

#define AS1 __attribute__((address_space(1)))
#define AS3 __attribute__((address_space(3)))

#if defined(__AMDGCN__) && __has_builtin(__builtin_amdgcn_global_load_async_to_lds_b128)
#define CBOW_ASYNC 1
#else
#define CBOW_ASYNC 0
#endif

#if CBOW_ASYNC && __has_builtin(__builtin_amdgcn_s_wait_asynccnt)
#define CBOW_PIPE 1
#else
#define CBOW_PIPE 0
#endif

__device__ __forceinline__ float wave_sum32(float x) {
    // wave32 butterfly reduction
    for (int off = 16; off > 0; off >>= 1)
        x += __shfl_xor(x, off, 32);
    return x;
}

__device__ __forceinline__ float dot4(v4f a, v4f b) {
    return a.x * b.x + a.y * b.y + a.z * b.z + a.w * b.w;
}

__device__ __forceinline__ float log_sigmoid(float x) {
    // numerically stable: min(x,0) - log1p(exp(-|x|))
    return fminf(x, 0.0f) - log1pf(__expf(-fabsf(x)));
}

__global__ __launch_bounds__(TPB) void cbow_main(
    const int*   __restrict__ ctx_words,
    const int*   __restrict__ tgt_words,
    const int*   __restrict__ neg_words,
    const float* __restrict__ V_emb,
    const float* __restrict__ U_emb,
    const float* __restrict__ mask_v,
    const float* __restrict__ mask_u,
    const float* __restrict__ mask_neg,
    float*       __restrict__ partials)
{
    const unsigned lane = threadIdx.x & 31u;
    const unsigned wv   = threadIdx.x >> 5;
    const unsigned b    = blockIdx.x * (unsigned)WPB + wv;  // one wave per batch

#if CBOW_ASYNC
    __shared__ v4f stage[WPB][CBOW_K][32];    // 40KB: 10 gathered 512B rows/wave
#endif
    __shared__ float wsum[WPB];

    const unsigned ctx = (unsigned)ctx_words[b];
    const unsigned tgt = (unsigned)tgt_words[b];

#if CBOW_ASYNC
    // Issue all 10 negative-row gathers up front: each instruction moves one
    // full 512B embedding row (b128 per lane x 32 lanes) global->LDS with no
    // VGPR return, tracked by ASYNCcnt.
#pragma unroll
    for (int k = 0; k < CBOW_K; ++k) {
        const unsigned nw = (unsigned)neg_words[b * (unsigned)CBOW_K + (unsigned)k];
        const float* gp = U_emb + (nw * CBOW_E + lane * 4u);
        __builtin_amdgcn_global_load_async_to_lds_b128(
            (AS1 v4i*)gp, (AS3 v4i*)(&stage[wv][k][lane]), 0, 0);
    }
#endif

    // Context row: regular (RT) load -> table stays L2-resident.
    // Mask rows: read exactly once -> non-temporal, don't pollute L2.
    const v4f vv = ((const v4f*)V_emb)[ctx * (CBOW_E / 4u) + lane];
    const v4f mv = __builtin_nontemporal_load(
        ((const v4f*)mask_v) + (b * (CBOW_E / 4u) + lane));
    const v4f vm = vv * mv;

    const v4f uu = ((const v4f*)U_emb)[tgt * (CBOW_E / 4u) + lane];
    const v4f mu = __builtin_nontemporal_load(
        ((const v4f*)mask_u) + (b * (CBOW_E / 4u) + lane));
    float pos = dot4(uu * mu, vm);

    // Stream all 10 negative-mask rows (NT) into registers first so the wait
    // pipeline below can't block their issue.
    v4f mn[CBOW_K];
    {
        const unsigned base = b * (unsigned)CBOW_K * (CBOW_E / 4u) + lane;
#pragma unroll
        for (int k = 0; k < CBOW_K; ++k)
            mn[k] = __builtin_nontemporal_load(
                ((const v4f*)mask_neg) + (base + (unsigned)k * (CBOW_E / 4u)));
    }

    float nacc = 0.0f;
#if CBOW_ASYNC
#if CBOW_PIPE
    // Async loads complete in order: consume row k after ASYNCcnt <= 9-k.
    // Manual unroll: the wait count must be a frontend constant expression.
#define CBOW_CONSUME(k, cnt)                                       \
    do {                                                           \
        __builtin_amdgcn_s_wait_asynccnt(cnt);                     \
        asm volatile("" ::: "memory");                             \
        const v4f un = stage[wv][(k)][lane];                       \
        nacc += dot4(un * mn[(k)], vm);                            \
    } while (0)
    CBOW_CONSUME(0, 9);
    CBOW_CONSUME(1, 8);
    CBOW_CONSUME(2, 7);
    CBOW_CONSUME(3, 6);
    CBOW_CONSUME(4, 5);
    CBOW_CONSUME(5, 4);
    CBOW_CONSUME(6, 3);
    CBOW_CONSUME(7, 2);
    CBOW_CONSUME(8, 1);
    CBOW_CONSUME(9, 0);
#undef CBOW_CONSUME
#else
    asm volatile("s_wait_asynccnt 0" ::: "memory");
#pragma unroll
    for (int k = 0; k < CBOW_K; ++k) {
        const v4f un = stage[wv][k][lane];
        nacc += dot4(un * mn[k], vm);
    }
#endif
#else
#pragma unroll
    for (int k = 0; k < CBOW_K; ++k) {
        const unsigned nw = (unsigned)neg_words[b * (unsigned)CBOW_K + (unsigned)k];
        const v4f un = ((const v4f*)U_emb)[nw * (CBOW_E / 4u) + lane];
        nacc += dot4(un * mn[k], vm);
    }
#endif

    pos  = wave_sum32(pos);
    nacc = wave_sum32(nacc);

    // u_neg was defined as the NEGATED masked rows, so neg_score = -nacc.
    if (lane == 0)
        wsum[wv] = log_sigmoid(pos) + log_sigmoid(-nacc);
    __syncthreads();
    if (threadIdx.x == 0) {
        float t = 0.0f;
        for (int i = 0; i < WPB; ++i) t += wsum[i];
        partials[blockIdx.x] = t;   // deterministic per-block partial
    }
}

__global__ __launch_bounds__(256) void cbow_reduce(
    const float* __restrict__ partials, float* __restrict__ out)
{
    float s = 0.0f;
    for (int i = threadIdx.x; i < NBLK; i += 256)
        s += partials[i];
    s = wave_sum32(s);

    __shared__ float ws[256 / 32];
    const int lane = threadIdx.x & 31;
    const int wv   = threadIdx.x >> 5;
    if (lane == 0) ws[wv] = s;
    __syncthreads();
    if (threadIdx.x == 0) {
        float t = 0.0f;
        for (int i = 0; i < 256 / 32; ++i) t += ws[i];
        out[0] = -t * (1.0f / (float)CBOW_B);   // -mean(loss)
    }
}

extern "C" void kernel_launch(void* const* d_in, const int* in_sizes, int n_in,
                              void* d_out, int out_size, void* d_ws, size_t ws_size,
                              hipStream_t stream) {
    const int*   ctx   = (const int*)  d_in[0];  // ctx_words    [B,1]
    const int*   tgt   = (const int*)  d_in[1];  // target_words [B,1]
    const int*   neg   = (const int*)  d_in[2];  // neg_words    [B,K]
    const float* V_emb = (const float*)d_in[3];  // [VOCAB,E]
    const float* U_emb = (const float*)d_in[4];  // [VOCAB,E]
    const float* mv    = (const float*)d_in[5];  // mask_v   [B,1,E]
    const float* mu    = (const float*)d_in[6];  // mask_u   [B,1,E]
    const float* mn    = (const float*)d_in[7];  // mask_neg [B,K,E]

    float* partials = (float*)d_ws;              // 8192 floats = 32KB scratch

    cbow_main<<<NBLK, TPB, 0, stream>>>(ctx, tgt, neg, V_emb, U_emb, mv, mu, mn, partials);
    cbow_reduce<<<1, 256, 0, stream>>>(partials, (float*)d_out);

    (void)in_sizes; (void)n_in; (void)out_size; (void)ws_size;
}